// DeformableConv_64862596104301
// MI455X (gfx1250) — compile-verified
//
#include <hip/hip_runtime.h>
#include <hip/hip_bf16.h>

// ---------------------------------------------------------------------------
// Deformable Conv v2 (B=8, C=128, H=W=80, K=3, pad=1) + BN + SiLU for gfx1250.
// fp32 WMMA (V_WMMA_F32_16X16X4_F32) everywhere; 2x2 C-tiling per wave;
// branchless clamp+select im2col fetches; sampled panel staged in LDS.
// ---------------------------------------------------------------------------

typedef float v2f __attribute__((ext_vector_type(2)));
typedef float v8f __attribute__((ext_vector_type(8)));

#define HH 80
#define WW 80
#define CIN 128
#define COUT 128
#define NPIX (HH * WW)            // 6400
#define RDIM (CIN * 9)            // 1152 reduction length (r = c*9 + tap)
#define PANEL (RDIM * 16)         // 18432 floats: one 16-pixel im2col panel
#define BTILES 200                // 32-pixel (2 rows x 16 cols) tiles per image
#define MAXOFF 20.0f              // max(H,W)/4

// ---------------------------------------------------------------------------
// Kernel 0: pack [w_off;w_mask;0] into 32x1152 matrix; fold BN into scale/shift
// ---------------------------------------------------------------------------
__global__ void __launch_bounds__(256) prep_kernel(
    const float* __restrict__ w_off, const float* __restrict__ w_mask,
    const float* __restrict__ gamma, const float* __restrict__ beta,
    const float* __restrict__ mean, const float* __restrict__ var,
    float* __restrict__ Wom, float* __restrict__ bnscale, float* __restrict__ bnshift) {
  int idx = blockIdx.x * 256 + threadIdx.x;
  if (idx < 32 * RDIM) {
    int row = idx / RDIM, r = idx - row * RDIM;
    float v = 0.0f;
    if (row < 18)      v = w_off[row * RDIM + r];        // OIHW flat == [o][c*9+t]
    else if (row < 27) v = w_mask[(row - 18) * RDIM + r];
    Wom[idx] = v;
  } else if (idx < 32 * RDIM + COUT) {
    int c = idx - 32 * RDIM;
    float s = gamma[c] * rsqrtf(var[c] + 1e-5f);
    bnscale[c] = s;
    bnshift[c] = beta[c] - mean[c] * s;
  }
}

// Branchless zero-padded im2col fetch for two adjacent output rows sharing
// one (channel, tap) decode. ym1 = h0 - 1 (top row y-base after padding).
__device__ __forceinline__ void im2col_fetch2(const float* __restrict__ xb,
                                              int cbase, int t, int ym1, int xcol,
                                              float& v0, float& v1) {
  int ki = (t >= 6) ? 2 : ((t >= 3) ? 1 : 0);
  int kj = t - 3 * ki;
  int xx = xcol + kj;
  bool okx = (unsigned)xx < (unsigned)WW;
  int ix = min(max(xx, 0), WW - 1);
  int yy0 = ym1 + ki;
  int yy1 = yy0 + 1;
  bool ok0 = okx && ((unsigned)yy0 < (unsigned)HH);
  bool ok1 = okx && ((unsigned)yy1 < (unsigned)HH);
  int iy0 = min(max(yy0, 0), HH - 1);
  int iy1 = min(max(yy1, 0), HH - 1);
  float a = xb[cbase + iy0 * WW + ix];   // unconditional load (clamped addr)
  float b = xb[cbase + iy1 * WW + ix];
  v0 = ok0 ? a : 0.0f;
  v1 = ok1 ? b : 0.0f;
}

// ---------------------------------------------------------------------------
// Kernel 1: offset/mask conv. One wave = 32 pixels (2 rows x 16) x 32 rows:
// 2x2 WMMA tiles sharing fetched operands. Epilogue: clip rows 0..17,
// 2*sigmoid rows 18..26, drop padding rows 27..31.
// ---------------------------------------------------------------------------
__global__ void __launch_bounds__(256) offmask_conv_kernel(
    const float* __restrict__ x, const float* __restrict__ Wom,
    float* __restrict__ om) {
  const int lane = threadIdx.x & 31;
  const int wv   = threadIdx.x >> 5;
  const int wtile = blockIdx.x * 8 + wv;        // 0..1599
  const int b   = wtile / BTILES;
  const int rem = wtile - b * BTILES;
  const int h0  = (rem / 5) * 2;                // rows h0, h0+1
  const int w0  = (rem - (rem / 5) * 5) * 16;

  const int n    = lane & 15;                   // pixel col within tile / N idx
  const int koff = (lane >> 4) << 1;            // 0 or 2 (K half selector)
  const float* xb = x + b * (CIN * NPIX);
  const float2* W2 = (const float2*)Wom;

  v8f acc00 = {}, acc01 = {}, acc10 = {}, acc11 = {};
  int cbase = 0, t = koff;                      // r = c*9 + t; cbase = c*NPIX
  const int xcol = w0 + n - 1;
  const int ym1  = h0 - 1;

#pragma unroll 2
  for (int r0 = 0; r0 < RDIM; r0 += 4) {
    int t1 = t + 1;
    int cbase1 = cbase;
    if (t1 == 9) { t1 = 0; cbase1 = cbase + NPIX; }

    float b00, b10, b01, b11;
    im2col_fetch2(xb, cbase,  t,  ym1, xcol, b00, b10);
    im2col_fetch2(xb, cbase1, t1, ym1, xcol, b01, b11);
    v2f bf0 = { b00, b01 };                     // pixel row h0
    v2f bf1 = { b10, b11 };                     // pixel row h0+1

    int acol = (r0 + koff) >> 1;
    float2 a0 = W2[(lane & 15) * (RDIM / 2) + acol];
    float2 a1 = W2[((lane & 15) + 16) * (RDIM / 2) + acol];
    v2f af0 = { a0.x, a0.y };
    v2f af1 = { a1.x, a1.y };

    acc00 = __builtin_amdgcn_wmma_f32_16x16x4_f32(false, af0, false, bf0, (short)0, acc00, false, false);
    acc01 = __builtin_amdgcn_wmma_f32_16x16x4_f32(false, af0, false, bf1, (short)0, acc01, false, false);
    acc10 = __builtin_amdgcn_wmma_f32_16x16x4_f32(false, af1, false, bf0, (short)0, acc10, false, false);
    acc11 = __builtin_amdgcn_wmma_f32_16x16x4_f32(false, af1, false, bf1, (short)0, acc11, false, false);

    t += 4;
    if (t >= 9) { t -= 9; cbase += NPIX; }
  }

  float* omb = om + b * (27 * NPIX);
  const int mhi = (lane >> 4) * 8;
  const int pix0 = h0 * WW + w0 + n;
  const int pix1 = pix0 + WW;
#pragma unroll
  for (int i = 0; i < 8; ++i) {
    int oc = i + mhi;                           // rows 0..15: all offset chans
    omb[oc * NPIX + pix0] = fminf(fmaxf(acc00[i], -MAXOFF), MAXOFF);
    omb[oc * NPIX + pix1] = fminf(fmaxf(acc01[i], -MAXOFF), MAXOFF);
  }
#pragma unroll
  for (int i = 0; i < 8; ++i) {
    int oc = 16 + i + mhi;                      // rows 16..31
    if (oc < 18) {
      omb[oc * NPIX + pix0] = fminf(fmaxf(acc10[i], -MAXOFF), MAXOFF);
      omb[oc * NPIX + pix1] = fminf(fmaxf(acc11[i], -MAXOFF), MAXOFF);
    } else if (oc < 27) {
      omb[oc * NPIX + pix0] = 2.0f / (1.0f + __expf(-acc10[i]));
      omb[oc * NPIX + pix1] = 2.0f / (1.0f + __expf(-acc11[i]));
    }
  }
}

// ---------------------------------------------------------------------------
// Kernel 2: deformable sampling + 128x1152 GEMM + BN + SiLU.
// Block = 32 pixels (2 rows x 16) x 128 oc; 8 waves, each owns two 16x16
// C-tiles (one per pixel row) sharing the A weight fragment.
// LDS: two stride-16 panels samp[2][1152][16] + 288 corner slots.
// ---------------------------------------------------------------------------
__global__ void __launch_bounds__(256) deform_gemm_kernel(
    const float* __restrict__ x, const float* __restrict__ w_main,
    const float* __restrict__ om, const float* __restrict__ bnscale,
    const float* __restrict__ bnshift, float* __restrict__ out) {
  extern __shared__ float lds[];
  float* samp = lds;                            // 2*PANEL floats
  float* wtab = lds + 2 * PANEL;                // 288*4 floats
  int*   itab = (int*)(wtab + 288 * 4);         // 288*4 ints

  const int tile = blockIdx.x;                  // 0..1599
  const int b   = tile / BTILES;
  const int rem = tile - b * BTILES;
  const int h0  = (rem / 5) * 2;
  const int w0  = (rem - (rem / 5) * 5) * 16;
  const float* xb  = x + b * (CIN * NPIX);
  const float* omb = om + b * (27 * NPIX);
  const int tid = threadIdx.x;

  // ---- phase 0: bilinear corner weights/indices per (tap, pixel) slot ----
  for (int j = tid; j < 288; j += 256) {
    int t = j >> 5;                             // tap 0..8
    int n = j & 31;                             // pixel 0..31
    int hr = h0 + (n >> 4);
    int wc = w0 + (n & 15);
    int pix = hr * WW + wc;
    float dy = omb[(2 * t) * NPIX + pix];
    float dx = omb[(2 * t + 1) * NPIX + pix];
    float m  = omb[(18 + t) * NPIX + pix];
    int ki = (t >= 6) ? 2 : ((t >= 3) ? 1 : 0);
    int kj = t - 3 * ki;
    float py = (float)(hr - 1 + ki) + dy;
    float px = (float)(wc - 1 + kj) + dx;
    float y0f = floorf(py), x0f = floorf(px);
    float wy = py - y0f, wx = px - x0f;
    int y0 = (int)y0f, x0 = (int)x0f;
    int y1 = y0 + 1,  x1 = x0 + 1;
    float vy0 = ((unsigned)y0 < (unsigned)HH) ? 1.0f : 0.0f;
    float vy1 = ((unsigned)y1 < (unsigned)HH) ? 1.0f : 0.0f;
    float vx0 = ((unsigned)x0 < (unsigned)WW) ? 1.0f : 0.0f;
    float vx1 = ((unsigned)x1 < (unsigned)WW) ? 1.0f : 0.0f;
    wtab[j * 4 + 0] = (1.0f - wy) * (1.0f - wx) * m * vy0 * vx0;
    wtab[j * 4 + 1] = (1.0f - wy) * wx          * m * vy0 * vx1;
    wtab[j * 4 + 2] = wy * (1.0f - wx)          * m * vy1 * vx0;
    wtab[j * 4 + 3] = wy * wx                   * m * vy1 * vx1;
    int y0c = min(max(y0, 0), HH - 1), y1c = min(max(y1, 0), HH - 1);
    int x0c = min(max(x0, 0), WW - 1), x1c = min(max(x1, 0), WW - 1);
    itab[j * 4 + 0] = y0c * WW + x0c;
    itab[j * 4 + 1] = y0c * WW + x1c;
    itab[j * 4 + 2] = y1c * WW + x0c;
    itab[j * 4 + 3] = y1c * WW + x1c;
  }
  __syncthreads();

  // ---- phase 1: samp[p][r][n16] = masked bilinear sample, r = c*9 + t ----
  // s enumerates panel0 then panel1; boundary (72*256) is wavefront-uniform.
  for (int s = tid; s < 2 * PANEL; s += 256) {
    int p  = (s >= PANEL) ? 1 : 0;
    int s2 = s - p * PANEL;
    int r  = s2 >> 4;
    int n16 = s2 & 15;
    unsigned c = (unsigned)r / 9u;
    int t = r - (int)c * 9;
    int j = (t << 5) + (p << 4) + n16;
    float4 wq = ((const float4*)wtab)[j];
    int4   iq = ((const int4*)itab)[j];
    const float* xc = xb + c * NPIX;
    samp[s] = wq.x * xc[iq.x] + wq.y * xc[iq.y] +
              wq.z * xc[iq.z] + wq.w * xc[iq.w];
  }
  __syncthreads();

  // ---- phase 2: per-wave 2x(16x16) C tiles, K=1152 in steps of 4 ----
  const int lane = tid & 31;
  const int wv   = tid >> 5;                    // oc tile 0..7
  const int n16  = lane & 15;
  const int koff = (lane >> 4) << 1;            // 0 or 2
  const int arow = wv * 16 + (lane & 15);
  const float2* Wm2 = (const float2*)w_main;    // [128][1152] row-major
  const float* samp0 = samp;
  const float* samp1 = samp + PANEL;
  v8f acc0 = {}, acc1 = {};
#pragma unroll 8
  for (int r0 = 0; r0 < RDIM; r0 += 4) {
    float2 a = Wm2[arow * (RDIM / 2) + ((r0 + koff) >> 1)];
    v2f af = { a.x, a.y };
    int bi = (r0 + koff) * 16 + n16;
    v2f bf0 = { samp0[bi], samp0[bi + 16] };
    v2f bf1 = { samp1[bi], samp1[bi + 16] };
    acc0 = __builtin_amdgcn_wmma_f32_16x16x4_f32(false, af, false, bf0, (short)0, acc0, false, false);
    acc1 = __builtin_amdgcn_wmma_f32_16x16x4_f32(false, af, false, bf1, (short)0, acc1, false, false);
  }

  // ---- fused BN + SiLU epilogue for both pixel rows ----
  const int pix0 = h0 * WW + w0 + n16;
  const int pix1 = pix0 + WW;
  const int mhi = (lane >> 4) * 8;
  float* outb = out + b * (COUT * NPIX);
#pragma unroll
  for (int i = 0; i < 8; ++i) {
    int oc = wv * 16 + i + mhi;
    float sc = bnscale[oc], sh = bnshift[oc];
    float ya = acc0[i] * sc + sh;
    float yb = acc1[i] * sc + sh;
    outb[oc * NPIX + pix0] = ya / (1.0f + __expf(-ya));
    outb[oc * NPIX + pix1] = yb / (1.0f + __expf(-yb));
  }
}

// ---------------------------------------------------------------------------
extern "C" void kernel_launch(void* const* d_in, const int* in_sizes, int n_in,
                              void* d_out, int out_size, void* d_ws, size_t ws_size,
                              hipStream_t stream) {
  const float* x      = (const float*)d_in[0];
  const float* w_main = (const float*)d_in[1];
  const float* w_off  = (const float*)d_in[2];
  const float* w_mask = (const float*)d_in[3];
  const float* gamma  = (const float*)d_in[4];
  const float* beta   = (const float*)d_in[5];
  const float* mean   = (const float*)d_in[6];
  const float* var    = (const float*)d_in[7];

  // Workspace (floats): Wom[32*1152] | bnscale[128] | bnshift[128] | om[8*27*6400]
  float* Wom     = (float*)d_ws;
  float* bnscale = Wom + 32 * RDIM;
  float* bnshift = bnscale + COUT;
  float* om      = bnshift + COUT;   // 5.53 MB

  int prep_elems = 32 * RDIM + COUT;
  prep_kernel<<<(prep_elems + 255) / 256, 256, 0, stream>>>(
      w_off, w_mask, gamma, beta, mean, var, Wom, bnscale, bnshift);

  // 1600 wave-tiles of 32 pixels; 8 waves per block -> 200 blocks
  offmask_conv_kernel<<<200, 256, 0, stream>>>(x, Wom, om);

  // one block per 32-pixel tile; dynamic LDS = 2 panels + corner tables
  size_t lds_bytes = (size_t)(2 * PANEL + 288 * 4 + 288 * 4) * 4;  // 156672 B
  (void)hipFuncSetAttribute((const void*)deform_gemm_kernel,
                            hipFuncAttributeMaxDynamicSharedMemorySize,
                            (int)lds_bytes);
  deform_gemm_kernel<<<1600, 256, lds_bytes, stream>>>(
      x, w_main, om, bnscale, bnshift, (float*)d_out);
}